// PNASimple_48137993454073
// MI455X (gfx1250) — compile-verified
//
#include <hip/hip_runtime.h>

#define F_DIM 48
#define EPS_F 1e-5f
#define AVG_LOG 2.8332133441f  // log(17.0)

typedef __attribute__((ext_vector_type(2))) float v2f;
typedef __attribute__((ext_vector_type(4))) float v4f;
typedef __attribute__((ext_vector_type(8))) float v8f;

// Order-preserving float -> uint key (monotone): enables exact min/max via u32 atomics.
__device__ __forceinline__ unsigned f2key(float f) {
    unsigned u = __float_as_uint(f);
    return (u & 0x80000000u) ? ~u : (u | 0x80000000u);
}
__device__ __forceinline__ float key2f(unsigned k) {
    unsigned u = (k & 0x80000000u) ? (k ^ 0x80000000u) : ~k;
    return __uint_as_float(u);
}

// ---------------- Kernel 1: init accumulators (must run every launch) ----------------
__global__ void __launch_bounds__(256) pna_init(float* __restrict__ sum,
                                                float* __restrict__ sq,
                                                unsigned* __restrict__ mnk,
                                                unsigned* __restrict__ mxk,
                                                float* __restrict__ deg,
                                                int nf, int n) {
    int i = blockIdx.x * 256 + threadIdx.x;
    if (i < nf) {
        sum[i] = 0.0f;
        sq[i]  = 0.0f;
        mnk[i] = 0xFFFFFFFFu;  // +inf key
        mxk[i] = 0x00000000u;  // -inf key
    }
    if (i < n) deg[i] = 0.0f;
}

// ---------------- Kernel 2: scatter with L2 atomics ----------------
// One thread per (edge, feature). x streamed non-temporally so the 77MB
// accumulator set stays resident in the 192MB L2 where atomics resolve.
__global__ void __launch_bounds__(256) pna_scatter(const float* __restrict__ x,
                                                   const int* __restrict__ index,
                                                   float* __restrict__ sum,
                                                   float* __restrict__ sq,
                                                   unsigned* __restrict__ mnk,
                                                   unsigned* __restrict__ mxk,
                                                   float* __restrict__ deg,
                                                   int total) {
    int i = blockIdx.x * 256 + threadIdx.x;
    if (i >= total) return;
    int e = i / F_DIM;
    int f = i - e * F_DIM;
    float v = __builtin_nontemporal_load(x + i);   // streaming: TH=NT
    int node = index[e];                           // shared by 48 threads: keep cached
    int o = node * F_DIM + f;
    atomicAdd(sum + o, v);
    atomicAdd(sq + o, v * v);
    unsigned key = f2key(v);
    atomicMin(mnk + o, key);
    atomicMax(mxk + o, key);
    if (f == 0) atomicAdd(deg + node, 1.0f);       // one per edge
}

// ---------------- Kernel 3: finalize with fp32 WMMA 16x16x4 ----------------
// Each wave produces 16 output rows. The 12-term dot (aggs x scalers) . w is
// done as 3 chained V_WMMA_F32_16X16X4_F32 (exact fp32 FMA, RNE).
__global__ void __launch_bounds__(256) pna_finalize(const float* __restrict__ sum,
                                                    const float* __restrict__ sq,
                                                    const unsigned* __restrict__ mnk,
                                                    const unsigned* __restrict__ mxk,
                                                    const float* __restrict__ deg,
                                                    const float* __restrict__ w,
                                                    float* __restrict__ out) {
    const int lane = threadIdx.x & 31;
    const int half = lane >> 4;                        // which K-pair this lane supplies
    const int wave = (blockIdx.x * blockDim.x + threadIdx.x) >> 5;
    const int R    = wave * 16;                        // 16 rows per wave, exact division
    const int row  = R + (lane & 15);                  // A-matrix row M = lane%16
    const int n    = row / F_DIM;

    float s  = sum[row];
    float q  = sq[row];
    float dg = deg[n];
    float safe = fmaxf(dg, 1.0f);
    float mean = s / safe;
    float var  = fmaxf(q / safe - mean * mean, 0.0f);
    float sd   = sqrtf(var + EPS_F);
    bool  ne   = dg > 0.0f;
    float mn   = ne ? key2f(mnk[row]) : 0.0f;          // empty segments -> 0 (torch_scatter)
    float mx   = ne ? key2f(mxk[row]) : 0.0f;
    float ld   = logf(dg + 1.0f);
    float amp  = ld * (1.0f / AVG_LOG);
    float att  = ne ? (AVG_LOG / ld) : 0.0f;

    // k = aggregator*3 + scaler; aggs = {mean, mn, mx, sd}; scalers = {1, amp, att}
    float k0 = mean,     k1 = mean * amp, k2  = mean * att;
    float k3 = mn,       k4 = mn   * amp, k5  = mn   * att;
    float k6 = mx,       k7 = mx   * amp, k8  = mx   * att;
    float k9 = sd,       k10 = sd  * amp, k11 = sd   * att;

    v8f c = {0.f, 0.f, 0.f, 0.f, 0.f, 0.f, 0.f, 0.f};
    {   // K group 0..3
        v2f a; a.x = half ? k2 : k0;  a.y = half ? k3 : k1;
        v2f b; b.x = w[half ? 2 : 0]; b.y = w[half ? 3 : 1];   // B[k][n] = w[k] for all n
        c = __builtin_amdgcn_wmma_f32_16x16x4_f32(false, a, false, b, (short)0, c, false, false);
    }
    {   // K group 4..7
        v2f a; a.x = half ? k6 : k4;  a.y = half ? k7 : k5;
        v2f b; b.x = w[half ? 6 : 4]; b.y = w[half ? 7 : 5];
        c = __builtin_amdgcn_wmma_f32_16x16x4_f32(false, a, false, b, (short)0, c, false, false);
    }
    {   // K group 8..11
        v2f a; a.x = half ? k10 : k8;  a.y = half ? k11 : k9;
        v2f b; b.x = w[half ? 10 : 8]; b.y = w[half ? 11 : 9];
        c = __builtin_amdgcn_wmma_f32_16x16x4_f32(false, a, false, b, (short)0, c, false, false);
    }

    // D columns are all identical; lanes 0 and 16 each hold rows R+0..7 / R+8..15
    if ((lane & 15) == 0) {
        float* o = out + R + 8 * half;
        v4f lo = {c[0], c[1], c[2], c[3]};
        v4f hi = {c[4], c[5], c[6], c[7]};
        __builtin_nontemporal_store(lo, (v4f*)o);
        __builtin_nontemporal_store(hi, (v4f*)(o + 4));
    }
}

extern "C" void kernel_launch(void* const* d_in, const int* in_sizes, int n_in,
                              void* d_out, int out_size, void* d_ws, size_t ws_size,
                              hipStream_t stream) {
    const float* x     = (const float*)d_in[0];   // [E, 48] f32
    const int*   index = (const int*)d_in[1];     // [E] i32
    // d_in[2] = dim_size (device scalar, derived from out_size instead)
    const float* w     = (const float*)d_in[3];   // [12] f32
    float* out = (float*)d_out;

    const int total = in_sizes[0];                // E * F = 76,800,000
    const int NF    = out_size;                   // N * F = 4,800,000
    const int N     = NF / F_DIM;                 // 100,000

    // Workspace layout: sum[NF] | sq[NF] | minkey[NF] | maxkey[NF] | deg[N]  (~77.2 MB)
    float*    sum = (float*)d_ws;
    float*    sq  = sum + NF;
    unsigned* mnk = (unsigned*)(sq + NF);
    unsigned* mxk = mnk + NF;
    float*    deg = (float*)(mxk + NF);

    pna_init<<<(NF + 255) / 256, 256, 0, stream>>>(sum, sq, mnk, mxk, deg, NF, N);
    pna_scatter<<<(total + 255) / 256, 256, 0, stream>>>(x, index, sum, sq, mnk, mxk, deg, total);

    const int waves  = NF / 16;        // 300,000 waves, 16 rows each — exact
    const int blocks = waves / 8;      // 8 waves (256 threads) per block — exact
    pna_finalize<<<blocks, 256, 0, stream>>>(sum, sq, mnk, mxk, deg, w, out);
}